// ReparamLogisticRegressionTorchModule_82025285419604
// MI455X (gfx1250) — compile-verified
//
#include <hip/hip_runtime.h>
#include <math.h>

// MI455X / gfx1250, wave32.
// mean = x @ w_mu ; var = (x*x) @ exp(10*w_log_var) ; out = std[:,None]*z + mean[:,None]
// Memory-bound: 204.8MB read + 51.2MB write ~= 11us floor @ 23.3TB/s.
// Phase 1: VALU streaming reduction (nontemporal b128 loads, LDS-broadcast weights).
// Phase 2: epilogue GEMM [N,2] x [2,64] via V_WMMA_F32_16X16X4_F32 (branch-free,
//          immediate-offset stores; EXEC stays all-ones through the WMMA region).

typedef __attribute__((ext_vector_type(2))) float v2f;
typedef __attribute__((ext_vector_type(4))) float v4f;
typedef __attribute__((ext_vector_type(8))) float v8f;

#define D_DIM 256
#define S_DIM 64

__global__ __launch_bounds__(256) void reparam_wmma_kernel(
    const float* __restrict__ x,
    const float* __restrict__ w_mu,
    const float* __restrict__ w_log_var,
    const float* __restrict__ z,
    float* __restrict__ out,
    int nRows)
{
    __shared__ __align__(16) float s_wmu[D_DIM];
    __shared__ __align__(16) float s_wexp[D_DIM];
    __shared__ __align__(16) float s_z[S_DIM];

    const int tid = threadIdx.x;
    // One element per thread (blockDim == D_DIM == 256)
    s_wmu[tid]  = w_mu[tid];
    s_wexp[tid] = __expf(10.0f * w_log_var[tid]);
    if (tid < S_DIM) s_z[tid] = z[tid];
    __syncthreads();

    const int lane = tid & 31;
    const int wave = tid >> 5;

    // Each wave owns 32 consecutive rows. nRows % 32 == 0 (200000), so the
    // guard is wave-uniform and surviving waves keep EXEC all-ones.
    const long long row0 = ((long long)blockIdx.x * 8 + wave) * 32;
    if (row0 >= (long long)nRows) return;

    const long long myRow = row0 + lane;
    const float* __restrict__ xr = x + myRow * D_DIM;

    // ---- Phase 1: dual dot-product over the row (streaming, b128 loads) ----
    float mean = 0.0f, var = 0.0f;
    #pragma unroll 8
    for (int k = 0; k < D_DIM; k += 4) {
        const v4f xv = __builtin_nontemporal_load((const v4f*)(xr + k));
        const v4f wm = *(const v4f*)(s_wmu + k);   // wave-uniform LDS broadcast
        const v4f we = *(const v4f*)(s_wexp + k);
        mean = fmaf(xv.x, wm.x, mean);
        mean = fmaf(xv.y, wm.y, mean);
        mean = fmaf(xv.z, wm.z, mean);
        mean = fmaf(xv.w, wm.w, mean);
        var  = fmaf(xv.x * we.x, xv.x, var);
        var  = fmaf(xv.y * we.y, xv.y, var);
        var  = fmaf(xv.z * we.z, xv.z, var);
        var  = fmaf(xv.w * we.w, xv.w, var);
    }
    const float stdv = sqrtf(var);

    // ---- Phase 2: out_tile = [std|mean] (16x2, zero-padded to 16x4) x [[z],[1]] (4x16) ----
    // A layout (16x4 f32): v0 = K0 (lanes 0-15) / K2 (lanes 16-31); v1 = K1 / K3.
    // B layout (4x16 f32): v0 = K0 row (lanes 0-15) / K2 (lanes 16-31); v1 = K1 / K3.
    const int halfLane = lane & 15;
    const bool loHalf  = (lane < 16);

    // Unconditional LDS broadcast reads (no exec-masked ds_load), then cndmask.
    float zv[4];
    #pragma unroll
    for (int jt = 0; jt < 4; ++jt) zv[jt] = s_z[jt * 16 + halfLane];

    // Single base pointer; every store below is a compile-time byte offset from it.
    float* __restrict__ op0 = out + (row0 + (loHalf ? 0 : 8)) * S_DIM + halfLane;

    #pragma unroll
    for (int t = 0; t < 2; ++t) {                 // two 16-row M tiles of this wave's 32 rows
        const float sA = __shfl(stdv, halfLane + t * 16, 32);
        const float mA = __shfl(mean, halfLane + t * 16, 32);
        v2f a;
        a.x = loHalf ? sA : 0.0f;                 // K=0 column: std  (K=2 -> 0)
        a.y = loHalf ? mA : 0.0f;                 // K=1 column: mean (K=3 -> 0)

        #pragma unroll
        for (int jt = 0; jt < 4; ++jt) {          // four 16-column N tiles of S=64
            v2f b;
            b.x = loHalf ? zv[jt] : 0.0f;         // K=0 row: z
            b.y = loHalf ? 1.0f  : 0.0f;          // K=1 row: ones

            v8f c = {};
            v8f d = __builtin_amdgcn_wmma_f32_16x16x4_f32(
                /*neg_a=*/false, a, /*neg_b=*/false, b,
                /*c_mod=*/(short)0, c, /*reuse_a=*/false, /*reuse_b=*/false);

            // D layout: VGPR r -> row (r | r+8), lane -> column within tile.
            // Offset = (t*16 + r) rows * 64 floats + jt*16 floats (all constants).
            #pragma unroll
            for (int r = 0; r < 8; ++r) {
                __builtin_nontemporal_store(
                    d[r], op0 + (t * 16 + r) * S_DIM + jt * 16);
            }
        }
    }
}

extern "C" void kernel_launch(void* const* d_in, const int* in_sizes, int n_in,
                              void* d_out, int out_size, void* d_ws, size_t ws_size,
                              hipStream_t stream) {
    const float* x     = (const float*)d_in[0];
    const float* w_mu  = (const float*)d_in[1];
    const float* w_lv  = (const float*)d_in[2];
    const float* z     = (const float*)d_in[3];
    float* out         = (float*)d_out;

    const int nRows = in_sizes[0] / D_DIM;           // 200000 (multiple of 32)
    const int rowsPerBlock = 256;                    // 8 waves x 32 rows
    const int grid = (nRows + rowsPerBlock - 1) / rowsPerBlock;

    reparam_wmma_kernel<<<grid, 256, 0, stream>>>(x, w_mu, w_lv, z, out, nRows);
}